// Graph_Encoder_1778116460939
// MI455X (gfx1250) — compile-verified
//
#include <hip/hip_runtime.h>

typedef float v2f __attribute__((ext_vector_type(2)));
typedef float v8f __attribute__((ext_vector_type(8)));

#define T_DIM 12
#define NS    100000
#define ND    49152
#define NE    200000
#define SH    9
#define H_DIM 128
#define PAD   24

// workspace layout (float offsets)
#define OFF_XFEAT 0
#define OFF_Q     (NS * PAD)                               // 2,400,000
#define OFF_NSRC  (OFF_Q + T_DIM * ND * PAD)               // Q: 14,155,776 floats
#define OFF_NDST  (OFF_NSRC + T_DIM * NS)                  // nsrc: 1,200,000 floats
#define ZERO_N    ((size_t)(T_DIM * ND * PAD + T_DIM * NS + T_DIM * ND))

// ---------------------------------------------------------------- zero scratch
__global__ __launch_bounds__(256) void k_zero(float* __restrict__ p, size_t n) {
  size_t i = (size_t)blockIdx.x * blockDim.x + threadIdx.x;
  size_t stride = (size_t)gridDim.x * blockDim.x;
  for (; i < n; i += stride) p[i] = 0.0f;
}

// ------------------------------------------- per-source-node 24-float feature
// xfeat[n][0..11]  = nan_to_num(x_src[t][n])
// xfeat[n][12..20] = pos_emb[emb_idx[n]][k] * pe_scale[k]
// xfeat[n][21..23] = 0 (padding so WMMA K pads are zero)
__global__ __launch_bounds__(256) void k_xfeat(const float* __restrict__ x,
                                               const float* __restrict__ pe,
                                               const float* __restrict__ scale,
                                               const int* __restrict__ eidx,
                                               float* __restrict__ xf) {
  int n = blockIdx.x * 256 + threadIdx.x;
  if (n >= NS) return;
  float* o = xf + (size_t)n * PAD;
#pragma unroll
  for (int t = 0; t < T_DIM; ++t) {
    float v = x[(size_t)t * NS + n];
    o[t] = (v == v) ? v : 0.0f;            // nan_to_num
  }
  int e = eidx[n];
#pragma unroll
  for (int k = 0; k < SH; ++k) o[12 + k] = pe[(size_t)e * SH + k] * scale[k];
  o[21] = 0.0f; o[22] = 0.0f; o[23] = 0.0f;
}

// ------------------------------------------------------------------- degrees
__global__ __launch_bounds__(256) void k_deg(const int* __restrict__ si,
                                             const int* __restrict__ di,
                                             float* __restrict__ dsrc,
                                             float* __restrict__ ddst) {
  int i = blockIdx.x * 256 + threadIdx.x;
  if (i >= T_DIM * NE) return;
  int t2 = i / NE;
  atomicAdd(&dsrc[(size_t)t2 * NS + si[i]], 1.0f);
  atomicAdd(&ddst[(size_t)t2 * ND + di[i]], 1.0f);
}

// deg -> clip(deg,1)^-0.5   (nsrc and ndst are contiguous in ws -> one pass)
__global__ __launch_bounds__(256) void k_norm(float* __restrict__ p, size_t n) {
  size_t i = (size_t)blockIdx.x * 256 + threadIdx.x;
  if (i >= n) return;
  p[i] = rsqrtf(fmaxf(p[i], 1.0f));
}

// --------------------------------------------------- edge scatter into Q
// Q[t2][d][j] += norm_src[t2][s] * xfeat[s][j],  j in [0,21); Q padded to 24.
// One lane per (edge,j): coalesced gather of xfeat, atomics land in one
// cacheline per edge.
__global__ __launch_bounds__(256) void k_scatter(const int* __restrict__ si,
                                                 const int* __restrict__ di,
                                                 const float* __restrict__ nsrc,
                                                 const float* __restrict__ xf,
                                                 float* __restrict__ Q) {
  unsigned tid = blockIdx.x * 256u + threadIdx.x;
  unsigned p = tid / PAD;
  unsigned j = tid % PAD;
  if (p >= (unsigned)(T_DIM * NE) || j >= 21u) return;
  unsigned t2 = p / NE;
  int s = si[p];
  int d = di[p];
  float w = nsrc[(size_t)t2 * NS + s];
  atomicAdd(&Q[((size_t)t2 * ND + d) * PAD + j], w * xf[(size_t)s * PAD + j]);
}

// ------------------------------------------------------- dense WMMA kernel
// Block: 16 dst rows (one M tile), 256 threads = 8 waves, wave w owns the
// 16-wide h tile [16w,16w+16). Per wave: 12 f32 16x16 accumulators (one per
// outer timestep t), loop over edge types t2:
//   C_P = WMMA^3( P[t2] 16x12 , W[t2][1..9] padded 12x16 )  (f32 16x16x4)
//   acc[t] += lrelu( nd * (A_t * W0 + C_P) + b )            (VALU on C layout)
__global__ __launch_bounds__(256) void k_dense(const float* __restrict__ Q,
                                               const float* __restrict__ ndst,
                                               const float* __restrict__ W,
                                               const float* __restrict__ B,
                                               float* __restrict__ out) {
  __shared__ float lA[T_DIM][16][12];   // A_t sums, t = 0..11
  __shared__ float lP[T_DIM][16][12];   // PE sums, k = 0..8 (+3 zero pad)
  __shared__ float lnd[T_DIM][16];      // dst norms

  const int tid = threadIdx.x;
  const int d0 = blockIdx.x * 16;

  for (int idx = tid; idx < T_DIM * 16 * PAD; idx += 256) {
    int t2 = idx / (16 * PAD);
    int rem = idx % (16 * PAD);
    int dl = rem / PAD, j = rem % PAD;
    float v = Q[((size_t)t2 * ND + d0 + dl) * PAD + j];
    if (j < 12) lA[t2][dl][j] = v;
    else        lP[t2][dl][j - 12] = v;      // j=21..23 are zero -> K pad
  }
  for (int idx = tid; idx < T_DIM * 16; idx += 256) {
    int t2 = idx / 16, dl = idx % 16;
    lnd[t2][dl] = ndst[(size_t)t2 * ND + d0 + dl];
  }
  __syncthreads();

  const int lane = tid & 31;           // wave32
  const int wv = tid >> 5;
  const int hl = lane & 15;            // N / column within tile
  const int hi = lane >> 4;            // half-wave select
  const int h = wv * 16 + hl;

  float acc[T_DIM][8];
#pragma unroll
  for (int t = 0; t < T_DIM; ++t)
#pragma unroll
    for (int r = 0; r < 8; ++r) acc[t][r] = 0.0f;

  for (int t2 = 0; t2 < T_DIM; ++t2) {
    const float* Wt = W + (size_t)t2 * (10 * H_DIM);
    v8f cP = {0, 0, 0, 0, 0, 0, 0, 0};
#pragma unroll
    for (int s = 0; s < 3; ++s) {
      // A 16x4 layout: lanes 0-15 & 16-31 both hold M=0..15; K split by half.
      int k0 = 4 * s + 2 * hi;
      v2f a, bm;
      a.x = lP[t2][hl][k0];
      a.y = lP[t2][hl][k0 + 1];
      // B 4x16 layout (mirrors A): row K in VGPR, split by half-wave.
      int row0 = 1 + k0;               // feat row 0 is the x-channel (A part)
      bm.x = (row0     < 10) ? Wt[row0 * H_DIM + h]       : 0.0f;
      bm.y = (row0 + 1 < 10) ? Wt[(row0 + 1) * H_DIM + h] : 0.0f;
      cP = __builtin_amdgcn_wmma_f32_16x16x4_f32(
          /*neg_a=*/false, a, /*neg_b=*/false, bm,
          /*c_mod=*/(short)0, cP, /*reuse_a=*/false, /*reuse_b=*/false);
    }
    const float w0 = Wt[h];                       // W[t2][0][h]
    const float bb = B[(size_t)t2 * H_DIM + h];   // b[t2][h]
#pragma unroll
    for (int r = 0; r < 8; ++r) {
      // C/D layout: lane 0-15 -> N=lane, VGPR r -> M=r (lanes 0-15) / r+8.
      int dl = r + 8 * hi;
      float ndv = lnd[t2][dl];
      float4 a0 = *(const float4*)&lA[t2][dl][0];
      float4 a1 = *(const float4*)&lA[t2][dl][4];
      float4 a2 = *(const float4*)&lA[t2][dl][8];
      float av[12] = {a0.x, a0.y, a0.z, a0.w, a1.x, a1.y, a1.z, a1.w,
                      a2.x, a2.y, a2.z, a2.w};
      float cpr = cP[r];
#pragma unroll
      for (int t = 0; t < T_DIM; ++t) {
        float z = fmaf(ndv, fmaf(av[t], w0, cpr), bb);
        z = (z > 0.0f) ? z : 0.01f * z;   // leaky_relu, slope 0.01
        acc[t][r] += z;
      }
    }
  }

  // out shape [Nd, 1, T, H] -> flat ((d*12)+t)*128 + h
#pragma unroll
  for (int r = 0; r < 8; ++r) {
    size_t d = (size_t)(d0 + r + 8 * hi);
#pragma unroll
    for (int t = 0; t < T_DIM; ++t) {
      out[(d * T_DIM + t) * H_DIM + h] = acc[t][r];
    }
  }
}

// ---------------------------------------------------------------------------
extern "C" void kernel_launch(void* const* d_in, const int* in_sizes, int n_in,
                              void* d_out, int out_size, void* d_ws, size_t ws_size,
                              hipStream_t stream) {
  const float* x_src    = (const float*)d_in[0];   // [12,100000,1]
  const float* pos_emb  = (const float*)d_in[1];   // [120000,9]
  const float* pe_scale = (const float*)d_in[2];   // [1,9]
  const int*   emb_idx  = (const int*)d_in[3];     // [100000]
  const int*   src_idx  = (const int*)d_in[4];     // [12,200000]
  const int*   dst_idx  = (const int*)d_in[5];     // [12,200000]
  const float* W        = (const float*)d_in[6];   // [12,10,128]
  const float* B        = (const float*)d_in[7];   // [12,128]
  float* out = (float*)d_out;                      // [49152,1,12,128]

  float* ws    = (float*)d_ws;
  float* xfeat = ws + OFF_XFEAT;
  float* Q     = ws + OFF_Q;
  float* nsrc  = ws + OFF_NSRC;
  float* ndst  = ws + OFF_NDST;

  // 1. zero Q + degree buffers (contiguous region)
  k_zero<<<8192, 256, 0, stream>>>(ws + OFF_Q, ZERO_N);
  // 2. per-node feature rows
  k_xfeat<<<(NS + 255) / 256, 256, 0, stream>>>(x_src, pos_emb, pe_scale,
                                                emb_idx, xfeat);
  // 3. degrees
  k_deg<<<(T_DIM * NE + 255) / 256, 256, 0, stream>>>(src_idx, dst_idx, nsrc,
                                                      ndst);
  // 4. norms (nsrc and ndst contiguous)
  k_norm<<<(T_DIM * (NS + ND) + 255) / 256, 256, 0, stream>>>(
      nsrc, (size_t)(T_DIM * (NS + ND)));
  // 5. edge scatter into Q
  {
    long long total = (long long)T_DIM * NE * PAD;   // 57,600,000
    int blocks = (int)((total + 255) / 256);
    k_scatter<<<blocks, 256, 0, stream>>>(src_idx, dst_idx, nsrc, xfeat, Q);
  }
  // 6. dense WMMA projection + leaky_relu + sum over edge types
  k_dense<<<ND / 16, 256, 0, stream>>>(Q, ndst, W, B, out);
}